// DualBranchGNN_33071248179680
// MI455X (gfx1250) — compile-verified
//
#include <hip/hip_runtime.h>
#include <math.h>

// Problem constants (match reference)
#define NN   50000
#define EE   1600000
#define IND  128
#define HH   64
#define SIG2X2 (2.0f * 30.0f * 30.0f)

typedef float v2f __attribute__((ext_vector_type(2)));
typedef float v8f __attribute__((ext_vector_type(8)));

__device__ __forceinline__ float atomicMaxF(float* addr, float val) {
  // IEEE-754 monotonic bit trick: works for any sign mix.
  if (val >= 0.0f)
    return __int_as_float(atomicMax((int*)addr, __float_as_int(val)));
  else
    return __uint_as_float(atomicMin((unsigned int*)addr, __float_as_uint(val)));
}

__device__ __forceinline__ float leaky02(float x) { return x > 0.0f ? x : 0.2f * x; }
__device__ __forceinline__ float elu1(float x)    { return x > 0.0f ? x : (__expf(x) - 1.0f); }

// ---------------------------------------------------------------------------
// K1: Y[N,64] = X[N,128] @ W[64,128]^T  via V_WMMA_F32_16X16X4_F32
// W staged in LDS once per block (32KB); one wave32 per 16-node row tile.
// A frag (16x4 f32): lane m=lane&15, ksel=2*(lane>>4) -> {A[m][k+ksel], A[m][k+ksel+1]}
// B frag (4x16 f32): B[k][n] = W[h0+n][k], same lane pattern with n=lane&15
// C/D (16x16 f32, 8 VGPRs): element (r + 8*(lane>>4), lane&15)
// ---------------------------------------------------------------------------
__global__ void __launch_bounds__(256) k_gemm_node(const float* __restrict__ X,
                                                   const float* __restrict__ W,
                                                   float* __restrict__ Y) {
  __shared__ float sW[HH * IND];                       // 32 KB
  for (int i = threadIdx.x; i < (HH * IND) / 4; i += blockDim.x)
    ((float4*)sW)[i] = ((const float4*)W)[i];          // global_load_b128 -> ds_store_b128
  __syncthreads();

  int wave = (blockIdx.x * blockDim.x + threadIdx.x) >> 5;
  int lane = threadIdx.x & 31;
  if (wave >= NN / 16) return;                         // wave-uniform: EXEC stays all-ones
  int row0 = wave * 16;
  int m    = lane & 15;
  int ksel = (lane >> 4) * 2;

  v8f c0 = {}, c1 = {}, c2 = {}, c3 = {};
  const float* xrow = X + (size_t)(row0 + m) * IND;
  const float* w0 = sW + ( 0 + m) * IND;
  const float* w1 = sW + (16 + m) * IND;
  const float* w2 = sW + (32 + m) * IND;
  const float* w3 = sW + (48 + m) * IND;

  for (int kk = 0; kk < IND; kk += 4) {
    int k = kk + ksel;
    v2f a;  a.x = xrow[k]; a.y = xrow[k + 1];
    v2f b0; b0.x = w0[k];  b0.y = w0[k + 1];           // ds_load_b64
    v2f b1; b1.x = w1[k];  b1.y = w1[k + 1];
    v2f b2; b2.x = w2[k];  b2.y = w2[k + 1];
    v2f b3; b3.x = w3[k];  b3.y = w3[k + 1];
    c0 = __builtin_amdgcn_wmma_f32_16x16x4_f32(false, a, false, b0, (short)0, c0, false, false);
    c1 = __builtin_amdgcn_wmma_f32_16x16x4_f32(false, a, false, b1, (short)0, c1, false, false);
    c2 = __builtin_amdgcn_wmma_f32_16x16x4_f32(false, a, false, b2, (short)0, c2, false, false);
    c3 = __builtin_amdgcn_wmma_f32_16x16x4_f32(false, a, false, b3, (short)0, c3, false, false);
  }

  int rbase = (lane >> 4) * 8;
  #pragma unroll
  for (int r = 0; r < 8; ++r) {
    size_t row = (size_t)(row0 + rbase + r) * HH;
    Y[row +  0 + m] = c0[r];
    Y[row + 16 + m] = c1[r];
    Y[row + 32 + m] = c2[r];
    Y[row + 48 + m] = c3[r];
  }
}

// ---------------------------------------------------------------------------
// K2: per-node prep: attention scalars, init mx (self-loop), den, deg, zero accums
// ---------------------------------------------------------------------------
__global__ void k_node_prep(const float* __restrict__ xl,
                            const float* __restrict__ att_s, const float* __restrict__ att_d,
                            float* __restrict__ a_src, float* __restrict__ a_dst,
                            float* __restrict__ mx, float* __restrict__ den,
                            float* __restrict__ deg,
                            float* __restrict__ gat_o, float* __restrict__ gcn_o) {
  int n = blockIdx.x * blockDim.x + threadIdx.x;
  if (n >= NN) return;
  const float* row = xl + (size_t)n * HH;
  float as = 0.0f, ad = 0.0f;
  #pragma unroll 8
  for (int h = 0; h < HH; ++h) { float v = row[h]; as += v * att_s[h]; ad += v * att_d[h]; }
  a_src[n] = as; a_dst[n] = ad;
  mx[n]  = leaky02(as + ad);   // self-loop always present & unmasked
  den[n] = 0.0f;
  deg[n] = 1.0f;               // self-loop weight 1
  float* go = gat_o + (size_t)n * HH;
  float* co = gcn_o + (size_t)n * HH;
  #pragma unroll 8
  for (int h = 0; h < HH; ++h) { go[h] = 0.0f; co[h] = 0.0f; }
}

// ---------------------------------------------------------------------------
// K3: per-edge noise MLP + gaussian weight; masked weight -> ew, deg, attn max
// ---------------------------------------------------------------------------
__global__ void k_edge_w(const int* __restrict__ ei,
                         const float* __restrict__ nf, const float* __restrict__ coord,
                         const float* __restrict__ fc1w, const float* __restrict__ fc1b,
                         const float* __restrict__ fc2w, const float* __restrict__ fc2b,
                         const float* __restrict__ a_src, const float* __restrict__ a_dst,
                         float* __restrict__ ew, float* __restrict__ deg, float* __restrict__ mx) {
  int e = blockIdx.x * blockDim.x + threadIdx.x;
  if (e >= EE) return;
  int s = ei[e], d = ei[EE + e];

  float x[10];
  #pragma unroll
  for (int k = 0; k < 10; ++k) x[k] = nf[s * 10 + k];
  float ni = fc2b[0];
  #pragma unroll
  for (int j = 0; j < 10; ++j) {
    float t = fc1b[j];
    #pragma unroll
    for (int k = 0; k < 10; ++k) t += x[k] * fc1w[j * 10 + k];
    ni += elu1(t) * fc2w[j];
  }

  float dx = coord[2 * s] - coord[2 * d];
  float dy = coord[2 * s + 1] - coord[2 * d + 1];
  float gw = __expf(-(dx * dx + dy * dy) / SIG2X2);
  float z  = gw * (1.0f + ni) - 1.0f;
  float w  = 0.1f + 1.9f / (1.0f + __expf(-z));

  if (w >= 0.2f) {
    ew[e] = w;
    atomicAdd(&deg[d], w);
    atomicMaxF(&mx[d], leaky02(a_src[s] + a_dst[d]));
  } else {
    ew[e] = 0.0f;  // w >= 0.1 always, so ew>0 <=> mask
  }
}

// ---------------------------------------------------------------------------
// K4: softmax denominator accumulation over masked edges
// ---------------------------------------------------------------------------
__global__ void k_edge_den(const int* __restrict__ ei, const float* __restrict__ ew,
                           const float* __restrict__ a_src, const float* __restrict__ a_dst,
                           const float* __restrict__ mx, float* __restrict__ den) {
  int e = blockIdx.x * blockDim.x + threadIdx.x;
  if (e >= EE) return;
  if (ew[e] <= 0.0f) return;
  int s = ei[e], d = ei[EE + e];
  float eatt = leaky02(a_src[s] + a_dst[d]);
  atomicAdd(&den[d], __expf(eatt - mx[d]));
}

// ---------------------------------------------------------------------------
// K5: self-loop contributions; finalizes den (den += exp(e_self - mx))
// ---------------------------------------------------------------------------
__global__ void k_node_self(const float* __restrict__ a_src, const float* __restrict__ a_dst,
                            const float* __restrict__ mx, float* __restrict__ den,
                            const float* __restrict__ deg,
                            const float* __restrict__ xl, const float* __restrict__ xg,
                            float* __restrict__ gat_o, float* __restrict__ gcn_o) {
  int n = blockIdx.x * blockDim.x + threadIdx.x;
  if (n >= NN) return;
  float eatt = leaky02(a_src[n] + a_dst[n]);
  float exs  = __expf(eatt - mx[n]);
  float dn   = den[n] + exs;
  den[n] = dn;
  float alpha = exs / dn;
  float norm  = 1.0f / deg[n];        // dis[n]*1*dis[n]
  const float* xls = xl + (size_t)n * HH;
  const float* xgs = xg + (size_t)n * HH;
  float* go = gat_o + (size_t)n * HH;
  float* co = gcn_o + (size_t)n * HH;
  #pragma unroll 8
  for (int h = 0; h < HH; ++h) {
    go[h] += alpha * xls[h];
    co[h] += norm * xgs[h];
  }
}

// ---------------------------------------------------------------------------
// K6: one wave32 per edge; scatter-add alpha*xl[s] and norm*xg[s] into dst rows
// ---------------------------------------------------------------------------
__global__ void k_edge_scatter(const int* __restrict__ ei, const float* __restrict__ ew,
                               const float* __restrict__ a_src, const float* __restrict__ a_dst,
                               const float* __restrict__ mx, const float* __restrict__ den,
                               const float* __restrict__ deg,
                               const float* __restrict__ xl, const float* __restrict__ xg,
                               float* __restrict__ gat_o, float* __restrict__ gcn_o) {
  int wid  = (blockIdx.x * blockDim.x + threadIdx.x) >> 5;
  int lane = threadIdx.x & 31;
  if (wid >= EE) return;
  float we = ew[wid];
  if (we <= 0.0f) return;
  int s = ei[wid], d = ei[EE + wid];
  float eatt  = leaky02(a_src[s] + a_dst[d]);
  float alpha = __expf(eatt - mx[d]) / den[d];
  float norm  = we * rsqrtf(deg[s] * deg[d]);
  const float* xls = xl + (size_t)s * HH;
  const float* xgs = xg + (size_t)s * HH;
  float* go = gat_o + (size_t)d * HH;
  float* co = gcn_o + (size_t)d * HH;
  atomicAdd(&go[lane],      alpha * xls[lane]);
  atomicAdd(&go[lane + 32], alpha * xls[lane + 32]);
  atomicAdd(&co[lane],      norm * xgs[lane]);
  atomicAdd(&co[lane + 32], norm * xgs[lane + 32]);
}

// ---------------------------------------------------------------------------
// K7: fusion layer 1 via WMMA: hid = relu([elu(gat+gb), relu(gcn+cb)] @ W1^T + b1)
// W1 staged in LDS; activations applied on-the-fly while building the A fragment.
// ---------------------------------------------------------------------------
__global__ void __launch_bounds__(256) k_fusion_gemm(const float* __restrict__ gat_o,
                                                     const float* __restrict__ gcn_o,
                                                     const float* __restrict__ gat_b,
                                                     const float* __restrict__ gcn_b,
                                                     const float* __restrict__ W1,
                                                     const float* __restrict__ b1,
                                                     float* __restrict__ hid) {
  __shared__ float sW[HH * IND];                       // 32 KB
  for (int i = threadIdx.x; i < (HH * IND) / 4; i += blockDim.x)
    ((float4*)sW)[i] = ((const float4*)W1)[i];
  __syncthreads();

  int wave = (blockIdx.x * blockDim.x + threadIdx.x) >> 5;
  int lane = threadIdx.x & 31;
  if (wave >= NN / 16) return;
  int row0 = wave * 16;
  int m    = lane & 15;
  int ksel = (lane >> 4) * 2;

  v8f c0 = {}, c1 = {}, c2 = {}, c3 = {};
  const float* grow = gat_o + (size_t)(row0 + m) * HH;
  const float* crow = gcn_o + (size_t)(row0 + m) * HH;
  const float* w0  = sW + ( 0 + m) * IND;
  const float* w1r = sW + (16 + m) * IND;
  const float* w2r = sW + (32 + m) * IND;
  const float* w3r = sW + (48 + m) * IND;

  for (int kk = 0; kk < IND; kk += 4) {
    int k = kk + ksel;
    v2f a;
    #pragma unroll
    for (int t = 0; t < 2; ++t) {
      int kt = k + t;
      float v;
      if (kt < HH) { v = elu1(grow[kt] + gat_b[kt]); }
      else         { int k2 = kt - HH; v = crow[k2] + gcn_b[k2]; v = v > 0.0f ? v : 0.0f; }
      if (t == 0) a.x = v; else a.y = v;
    }
    v2f b0;  b0.x  = w0[k];  b0.y  = w0[k + 1];
    v2f b1v; b1v.x = w1r[k]; b1v.y = w1r[k + 1];
    v2f b2;  b2.x  = w2r[k]; b2.y  = w2r[k + 1];
    v2f b3;  b3.x  = w3r[k]; b3.y  = w3r[k + 1];
    c0 = __builtin_amdgcn_wmma_f32_16x16x4_f32(false, a, false, b0,  (short)0, c0, false, false);
    c1 = __builtin_amdgcn_wmma_f32_16x16x4_f32(false, a, false, b1v, (short)0, c1, false, false);
    c2 = __builtin_amdgcn_wmma_f32_16x16x4_f32(false, a, false, b2,  (short)0, c2, false, false);
    c3 = __builtin_amdgcn_wmma_f32_16x16x4_f32(false, a, false, b3,  (short)0, c3, false, false);
  }

  int rbase = (lane >> 4) * 8;
  #pragma unroll
  for (int r = 0; r < 8; ++r) {
    size_t row = (size_t)(row0 + rbase + r) * HH;
    float v;
    v = c0[r] + b1[ 0 + m]; hid[row +  0 + m] = v > 0.0f ? v : 0.0f;
    v = c1[r] + b1[16 + m]; hid[row + 16 + m] = v > 0.0f ? v : 0.0f;
    v = c2[r] + b1[32 + m]; hid[row + 32 + m] = v > 0.0f ? v : 0.0f;
    v = c3[r] + b1[48 + m]; hid[row + 48 + m] = v > 0.0f ? v : 0.0f;
  }
}

// ---------------------------------------------------------------------------
// K8: output head: out[n] = hid[n,:] . w2 + b2
// ---------------------------------------------------------------------------
__global__ void k_out(const float* __restrict__ hid, const float* __restrict__ w2,
                      const float* __restrict__ b2, float* __restrict__ out) {
  int n = blockIdx.x * blockDim.x + threadIdx.x;
  if (n >= NN) return;
  const float* row = hid + (size_t)n * HH;
  float acc = b2[0];
  #pragma unroll 8
  for (int j = 0; j < HH; ++j) acc += row[j] * w2[j];
  out[n] = acc;
}

// ---------------------------------------------------------------------------
extern "C" void kernel_launch(void* const* d_in, const int* in_sizes, int n_in,
                              void* d_out, int out_size, void* d_ws, size_t ws_size,
                              hipStream_t stream) {
  const float* x_local  = (const float*)d_in[0];
  const float* x_global = (const float*)d_in[1];
  const float* nf       = (const float*)d_in[2];
  const float* coord    = (const float*)d_in[3];
  const int*   ei       = (const int*)d_in[4];
  const float* fc1w = (const float*)d_in[5];
  const float* fc1b = (const float*)d_in[6];
  const float* fc2w = (const float*)d_in[7];
  const float* fc2b = (const float*)d_in[8];
  const float* gat_w = (const float*)d_in[9];
  const float* att_s = (const float*)d_in[10];
  const float* att_d = (const float*)d_in[11];
  const float* gat_b = (const float*)d_in[12];
  const float* gcn_w = (const float*)d_in[13];
  const float* gcn_b = (const float*)d_in[14];
  const float* fw1  = (const float*)d_in[15];
  const float* fb1  = (const float*)d_in[16];
  const float* fw2  = (const float*)d_in[17];
  const float* fb2  = (const float*)d_in[18];

  float* ws = (float*)d_ws;
  float* xl    = ws; ws += (size_t)NN * HH;   // reused as hid after scatter
  float* xg    = ws; ws += (size_t)NN * HH;
  float* gat_o = ws; ws += (size_t)NN * HH;
  float* gcn_o = ws; ws += (size_t)NN * HH;
  float* a_src = ws; ws += NN;
  float* a_dst = ws; ws += NN;
  float* mx    = ws; ws += NN;
  float* den   = ws; ws += NN;
  float* deg   = ws; ws += NN;
  float* ew    = ws; ws += EE;

  const int gemm_waves  = NN / 16;                            // 3125
  const int gemm_blocks = (gemm_waves + 7) / 8;               // 8 waves / 256-thread block

  k_gemm_node<<<gemm_blocks, 256, 0, stream>>>(x_local, gat_w, xl);
  k_gemm_node<<<gemm_blocks, 256, 0, stream>>>(x_global, gcn_w, xg);

  k_node_prep<<<(NN + 255) / 256, 256, 0, stream>>>(xl, att_s, att_d, a_src, a_dst,
                                                    mx, den, deg, gat_o, gcn_o);

  k_edge_w<<<(EE + 255) / 256, 256, 0, stream>>>(ei, nf, coord, fc1w, fc1b, fc2w, fc2b,
                                                 a_src, a_dst, ew, deg, mx);

  k_edge_den<<<(EE + 255) / 256, 256, 0, stream>>>(ei, ew, a_src, a_dst, mx, den);

  k_node_self<<<(NN + 255) / 256, 256, 0, stream>>>(a_src, a_dst, mx, den, deg,
                                                    xl, xg, gat_o, gcn_o);

  // one wave32 per edge
  long long scat_threads = (long long)EE * 32;
  k_edge_scatter<<<(unsigned)((scat_threads + 255) / 256), 256, 0, stream>>>(
      ei, ew, a_src, a_dst, mx, den, deg, xl, xg, gat_o, gcn_o);

  float* hid = xl;  // xl dead after scatter; reuse
  k_fusion_gemm<<<gemm_blocks, 256, 0, stream>>>(gat_o, gcn_o, gat_b, gcn_b, fw1, fb1, hid);

  k_out<<<(NN + 255) / 256, 256, 0, stream>>>(hid, fw2, fb2, (float*)d_out);
}